// NattenBlock_52682068853109
// MI455X (gfx1250) — compile-verified
//
#include <hip/hip_runtime.h>
#include <math.h>
#include <stdint.h>

typedef float v2f __attribute__((ext_vector_type(2)));
typedef float v8f __attribute__((ext_vector_type(8)));

#define D_MODEL 768
#define QKV_DIM 1280
#define NTOK    2048          // 2 * 32 * 32 tokens
#define NHEAD   12
#define KSZ     7

// ---------------------------------------------------------------------------
// Wave-tiled fp32 GEMM:  C[M,N] = A[M,768] * B[N,768]^T
// One wave owns a 16x16 C tile; 4 waves per block share a 16x768 A strip
// staged in LDS via CDNA5 async global->LDS copies (ASYNCcnt-tracked, no VGPR
// round trip). Inner loop: ds_load_b64 (A frag) + global_load_b64 (B frag) +
// v_wmma_f32_16x16x4_f32 (exact fp32 matrix op on CDNA5), unrolled 8x.
// Fragment layouts per CDNA5 ISA 7.12.2:
//   A 16x4 : lane m=lane&15, elem v holds K = k0 + 2*(lane>>4) + v
//   B 4x16 : lane n=lane&15, elem v holds K = k0 + 2*(lane>>4) + v
//   C 16x16: reg r -> M = r + 8*(lane>>4), N = lane&15
// ---------------------------------------------------------------------------
__global__ __launch_bounds__(128) void gemm_f32_wmma(
    const float* __restrict__ A, const float* __restrict__ B,
    float* __restrict__ C, int N) {
  __shared__ float tileA[16 * D_MODEL];   // 48 KB

  const int mtile = blockIdx.x;

  // ---- async global->LDS staging of the 16x768 A strip (b128 per lane) ----
  const float4* Av = (const float4*)(A + (size_t)mtile * 16 * D_MODEL);
  for (int idx = threadIdx.x; idx < 16 * D_MODEL / 4; idx += blockDim.x) {
    const float4* gp = Av + idx;
    // low 32 bits of a flat shared-memory address are the LDS byte offset
    unsigned loff = (unsigned)(uintptr_t)(tileA + idx * 4);
    asm volatile("global_load_async_to_lds_b128 %0, %1, off"
                 :: "v"(loff), "v"(gp) : "memory");
  }
  asm volatile("s_wait_asynccnt 0x0" ::: "memory");
  __syncthreads();

  const int lane  = threadIdx.x & 31;
  const int wave  = threadIdx.x >> 5;
  const int ntile = blockIdx.y * 4 + wave;
  const int m     = lane & 15;
  const int chi   = lane >> 4;
  const int n     = lane & 15;
  const float* Brow = B + (size_t)(ntile * 16 + n) * D_MODEL;

  // hint the B row stream into cache (global_prefetch_b8)
  __builtin_prefetch(Brow, 0, 1);

  v8f acc = {};
#pragma unroll 8
  for (int k0 = 0; k0 < D_MODEL; k0 += 4) {
    const int c0 = k0 + 2 * chi;
    v2f a, bf;
    a.x  = tileA[m * D_MODEL + c0];
    a.y  = tileA[m * D_MODEL + c0 + 1];
    bf.x = Brow[c0];
    bf.y = Brow[c0 + 1];
    acc = __builtin_amdgcn_wmma_f32_16x16x4_f32(
        /*neg_a=*/false, a, /*neg_b=*/false, bf,
        /*c_mod=*/(short)0, acc, /*reuse_a=*/false, /*reuse_b=*/false);
  }

  float* Crow = C + (size_t)(mtile * 16) * N + ntile * 16 + n;
#pragma unroll
  for (int r = 0; r < 8; ++r) {
    const int mm = r + 8 * chi;
    Crow[(size_t)mm * N] = acc[r];
  }
}

// ---------------------------------------------------------------------------
// Wave-wide reductions (wave32)
// ---------------------------------------------------------------------------
__device__ __forceinline__ float wsum(float x) {
#pragma unroll
  for (int m = 16; m > 0; m >>= 1) x += __shfl_xor(x, m, 32);
  return x;
}
__device__ __forceinline__ float wmax(float x) {
#pragma unroll
  for (int m = 16; m > 0; m >>= 1) x = fmaxf(x, __shfl_xor(x, m, 32));
  return x;
}

// ---------------------------------------------------------------------------
// 7x7 clipped-window neighborhood attention, one wave per (batch, head, pixel).
// Lane l owns channels {l, l+32} of D_HEAD=64. GQA: q-head h uses kv-head h%4
// (jnp.tile semantics). qkv row layout: [q 0:768 | k 768:1024 | v 1024:1280].
// ---------------------------------------------------------------------------
__global__ __launch_bounds__(256) void natten_attn(
    const float* __restrict__ qkv, float* __restrict__ obuf) {
  const int lane = threadIdx.x & 31;
  const int g    = blockIdx.x * 8 + (threadIdx.x >> 5);
  const int b    = g / (NHEAD * 1024);
  const int rem  = g % (NHEAD * 1024);
  const int h    = rem / 1024;
  const int p    = rem % 1024;
  const int i    = p >> 5, j = p & 31;
  const int hk   = h & 3;                       // h % kv_heads(4)
  const int i0   = min(max(i - 3, 0), 25);      // clip(i-3, 0, 32-7)
  const int j0   = min(max(j - 3, 0), 25);

  const size_t tq = (size_t)(b * 1024 + p);
  const float scale = 0.125f;                   // 1/sqrt(64)
  const float q0 = qkv[tq * QKV_DIM + h * 64 + lane] * scale;
  const float q1 = qkv[tq * QKV_DIM + h * 64 + 32 + lane] * scale;

  // ---- logits: 49 wave-reduced dot products, stored 2-per-lane ----
  float lg0 = -INFINITY, lg1 = -INFINITY;
  for (int w = 0; w < KSZ * KSZ; ++w) {
    const int a = w / KSZ, bb = w % KSZ;
    const size_t tk = (size_t)(b * 1024 + (i0 + a) * 32 + (j0 + bb));
    const float* kr = qkv + tk * QKV_DIM + 768 + hk * 64;
    const float logit = wsum(q0 * kr[lane] + q1 * kr[lane + 32]);
    if (w < 32) { if (lane == w)      lg0 = logit; }
    else        { if (lane == w - 32) lg1 = logit; }
  }

  // ---- softmax over the 49 values ----
  const float mx = wmax(fmaxf(lg0, lg1));
  float e0 = __expf(lg0 - mx);
  float e1 = (lane <= 16) ? __expf(lg1 - mx) : 0.0f;
  const float inv = 1.0f / wsum(e0 + e1);
  e0 *= inv;
  e1 *= inv;

  // ---- AV: broadcast each weight, accumulate V ----
  float o0 = 0.0f, o1 = 0.0f;
  for (int w = 0; w < KSZ * KSZ; ++w) {
    const int a = w / KSZ, bb = w % KSZ;
    const size_t tk = (size_t)(b * 1024 + (i0 + a) * 32 + (j0 + bb));
    const float* vr = qkv + tk * QKV_DIM + 1024 + hk * 64;
    const float aw = (w < 32) ? __shfl(e0, w, 32) : __shfl(e1, w - 32, 32);
    o0 += aw * vr[lane];
    o1 += aw * vr[lane + 32];
  }
  obuf[tq * D_MODEL + h * 64 + lane]      = o0;
  obuf[tq * D_MODEL + h * 64 + 32 + lane] = o1;
}

// ---------------------------------------------------------------------------
extern "C" void kernel_launch(void* const* d_in, const int* in_sizes, int n_in,
                              void* d_out, int out_size, void* d_ws, size_t ws_size,
                              hipStream_t stream) {
  const float* x     = (const float*)d_in[0];   // (2,32,32,768)
  const float* w_qkv = (const float*)d_in[1];   // (1280,768)
  const float* w_out = (const float*)d_in[2];   // (768,768)
  float* out = (float*)d_out;                   // (2,32,32,768)

  float* qkv  = (float*)d_ws;                   // (2048,1280)
  float* obuf = qkv + (size_t)NTOK * QKV_DIM;   // (2048,768)

  // 1) QKV projection: (2048x768) @ (1280x768)^T -> (2048x1280)
  gemm_f32_wmma<<<dim3(NTOK / 16, QKV_DIM / 64), 128, 0, stream>>>(
      x, w_qkv, qkv, QKV_DIM);

  // 2) Neighborhood attention: 2*12*1024 waves, 8 waves/block
  natten_attn<<<dim3((2 * NHEAD * 1024) / 8), 256, 0, stream>>>(qkv, obuf);

  // 3) Output projection: (2048x768) @ (768x768)^T -> (2048x768)
  gemm_f32_wmma<<<dim3(NTOK / 16, D_MODEL / 64), 128, 0, stream>>>(
      obuf, w_out, out, D_MODEL);
}